// CompressionLayer_24618752541064
// MI455X (gfx1250) — compile-verified
//
#include <hip/hip_runtime.h>

// MI455X / gfx1250, wave32. Grouped GEMM via V_WMMA_F32_16X16X4_F32.
// Per-wave 16(M) x 32(O) tile: one A fragment feeds two WMMAs -> 1.5 LDS
// b64 loads per WMMA instead of 2.

typedef __attribute__((ext_vector_type(2))) float v2f;
typedef __attribute__((ext_vector_type(4))) float f4;
typedef __attribute__((ext_vector_type(8))) float v8f;

#define BATCH   256
#define HW      512
#define NCH     1024
#define KIN     256
#define KOUT    64
#define LDS_STR 260   // 256 + 4-float pad to break LDS bank conflicts

__global__ __launch_bounds__(256, 2)
void compression_wmma_kernel(const float* __restrict__ x,
                             const float* __restrict__ Wk,
                             const float* __restrict__ bk,
                             float* __restrict__ out)
{
    __shared__ float Xs[64 * LDS_STR];   // 64 batch rows x 256 k   (66.6 KB)
    __shared__ float Ws[64 * LDS_STR];   // 64 out ch   x 256 k     (66.6 KB)

    const int n   = blockIdx.x;          // chunk index 0..1023
    const int m0  = blockIdx.y << 6;     // batch tile base (64 rows)
    const int ci  = n >> 5;              // chunk row in H
    const int cj  = n & 31;              // chunk col in W
    const int tid = threadIdx.x;

    // ---- cooperative coalesced load: X tile [64][256] -> LDS ----
    // xc[b][n][k], k = kh*16 + kw maps to x[b][(ci*16+kh)*512 + cj*16 + kw]
    const float* xbase = x + (size_t)(ci * 16) * HW + (size_t)(cj * 16);
    #pragma unroll
    for (int ii = 0; ii < 16; ++ii) {
        int idx = tid + (ii << 8);           // 4096 float4 slots
        int r   = idx >> 6;                  // local batch row 0..63
        int k   = (idx & 63) << 2;           // k multiple of 4 (stays in one 16-seg)
        int kh  = k >> 4, kw = k & 15;
        const float* gp = xbase + (size_t)(m0 + r) * (HW * HW) + kh * HW + kw;
        f4 v = *(const f4*)gp;
        *(f4*)&Xs[r * LDS_STR + k] = v;
    }

    // ---- cooperative coalesced load: Wk[n] [64][256] -> LDS ----
    const float* wbase = Wk + (size_t)n * (KOUT * KIN);
    #pragma unroll
    for (int ii = 0; ii < 16; ++ii) {
        int idx = tid + (ii << 8);           // 4096 float4 slots
        int r   = idx >> 6;                  // o 0..63
        int k   = (idx & 63) << 2;
        f4 v = *(const f4*)(wbase + r * KIN + k);
        *(f4*)&Ws[r * LDS_STR + k] = v;
    }
    __syncthreads();

    // ---- per-wave 16x32 output tile, K-loop of 16x16x4 f32 WMMA ----
    const int wave  = tid >> 5;
    const int lane  = tid & 31;
    const int mloc  = (wave >> 1) << 4;      // local M tile base: 0/16/32/48
    const int o0    = (wave & 1) << 5;       // O tile pair base: 0/32
    const int lrow  = lane & 15;
    const int khalf = lane >> 4;             // lanes 0-15: K=0..1, lanes 16-31: K=2..3

    // A frag: lane holds X[m0+mloc+lrow][k0 + khalf*2 + {0,1}]
    const float* ax  = &Xs[(mloc + lrow) * LDS_STR + khalf * 2];
    // B frags: B[k][o] = Wk[n][o][k]; lane holds Ws[o+lrow][k0 + khalf*2 + {0,1}]
    const float* bx0 = &Ws[(o0 + lrow) * LDS_STR + khalf * 2];
    const float* bx1 = bx0 + 16 * LDS_STR;

    v8f acc0 = {};
    v8f acc1 = {};
    #pragma unroll 8
    for (int k0 = 0; k0 < KIN; k0 += 4) {
        v2f a  = *(const v2f*)(ax  + k0);
        v2f b0 = *(const v2f*)(bx0 + k0);
        v2f b1 = *(const v2f*)(bx1 + k0);
        // 8 args: (neg_a, A, neg_b, B, c_mod, C, reuse_a, reuse_b)
        acc0 = __builtin_amdgcn_wmma_f32_16x16x4_f32(
            false, a, false, b0, (short)0, acc0, false, false);
        acc1 = __builtin_amdgcn_wmma_f32_16x16x4_f32(
            false, a, false, b1, (short)0, acc1, false, false);
    }

    // ---- epilogue: bias + ReLU + scatter (recombine mapping) ----
    // C/D layout: VGPR v, lane L -> M = v + 8*(L>>4), N = L&15
    #pragma unroll
    for (int t = 0; t < 2; ++t) {
        const int   o    = o0 + (t << 4) + lrow;
        const float bias = bk[n * KOUT + o];
        const int   oh   = o >> 3, ow = o & 7;
        const int   rowbase = (ci * 8 + oh) * 256 + cj * 8 + ow;  // in [256x256]
        const v8f&  acc  = t ? acc1 : acc0;
        #pragma unroll
        for (int v = 0; v < 8; ++v) {
            int   m   = m0 + mloc + v + (khalf << 3);
            float val = acc[v] + bias;
            val = val > 0.0f ? val : 0.0f;
            out[(size_t)m * (256 * 256) + rowbase] = val;
        }
    }
}

extern "C" void kernel_launch(void* const* d_in, const int* in_sizes, int n_in,
                              void* d_out, int out_size, void* d_ws, size_t ws_size,
                              hipStream_t stream) {
    const float* x  = (const float*)d_in[0];   // [256, 512, 512] f32
    const float* Wk = (const float*)d_in[1];   // [1024, 64, 256] f32
    const float* bk = (const float*)d_in[2];   // [1024, 64]      f32
    float* out = (float*)d_out;                // [256, 65536]    f32

    dim3 grid(NCH, BATCH / 64);                // 1024 chunks x 4 batch tiles
    compression_wmma_kernel<<<grid, 256, 0, stream>>>(x, Wk, bk, out);
}